// GCNPairTwoConv_10024453669564
// MI455X (gfx1250) — compile-verified
//
#include <hip/hip_runtime.h>
#include <math.h>

// ---------------------------------------------------------------------------
// GCN pair model for MI455X (gfx1250, wave32).
// Dense GEMMs via V_WMMA_F32_16X16X4_F32 (fp32-exact matrix pipe);
// graph aggregation via warp-per-edge global_atomic_add_f32 (bandwidth bound).
// ---------------------------------------------------------------------------

typedef float v2f __attribute__((ext_vector_type(2)));
typedef float v8f __attribute__((ext_vector_type(8)));

#define TPB 256

__global__ void k_fill(float* __restrict__ p, float v, int n) {
  int i = blockIdx.x * blockDim.x + threadIdx.x;
  if (i < n) p[i] = v;
}

// Y[M,Kout] = act( opt_relu(X)[M,Kin] @ W[Kin,Kout] (+ bias) )
// One block per 16-row strip of X (staged in LDS, padded stride),
// one wave per 16x16 output tile, K stepped by 4 through fp32 WMMA.
__global__ void k_gemm_wmma(const float* __restrict__ X, const float* __restrict__ W,
                            const float* __restrict__ bias, float* __restrict__ Y,
                            int M, int Kin, int Kout,
                            int relu_in, int relu_out, int use_bias)
{
  extern __shared__ float lds[];
  const int ldl = Kin + 1;                 // pad to avoid 64-bank conflicts
  const int m0  = blockIdx.x * 16;

  for (int i = threadIdx.x; i < 16 * Kin; i += blockDim.x) {
    int r = i / Kin, c = i - r * Kin;
    int row = m0 + r;
    float v = (row < M) ? X[(size_t)row * Kin + c] : 0.0f;
    if (relu_in) v = fmaxf(v, 0.0f);
    lds[r * ldl + c] = v;
  }
  __syncthreads();

  const int wave = threadIdx.x >> 5;
  const int lane = threadIdx.x & 31;
  const int half = lane >> 4;              // 0: lanes 0-15, 1: lanes 16-31
  const int l    = lane & 15;
  const int n0   = wave * 16;
  if (n0 >= Kout) return;

  v8f acc = {};
  for (int k = 0; k < Kin; k += 4) {
    const int ka = k + 2 * half;           // A 16x4 layout: half selects K pair
    v2f a, b;
    a.x = lds[l * ldl + ka];
    a.y = lds[l * ldl + ka + 1];
    b.x = W[(size_t)ka * Kout + n0 + l];   // B 4x16: rows striped across lanes
    b.y = W[(size_t)(ka + 1) * Kout + n0 + l];
    acc = __builtin_amdgcn_wmma_f32_16x16x4_f32(
        /*neg_a=*/false, a, /*neg_b=*/false, b,
        /*c_mod=*/(short)0, acc, /*reuse_a=*/false, /*reuse_b=*/false);
  }

  float bb = use_bias ? bias[n0 + l] : 0.0f;
#pragma unroll
  for (int i = 0; i < 8; ++i) {
    int row = m0 + i + 8 * half;           // C/D layout: VGPR i -> M=i / M=8+i
    if (row < M) {
      float v = acc[i] + bb;
      if (relu_out) v = fmaxf(v, 0.0f);
      Y[(size_t)row * Kout + n0 + l] = v;
    }
  }
}

// g[n] = dot(Hid[n, 0:64], W2) + b2   (warp per node, F=64)
__global__ void k_gate_g(const float* __restrict__ Hid, const float* __restrict__ W2,
                         const float* __restrict__ b2, float* __restrict__ g, int n) {
  int w = (blockIdx.x * blockDim.x + threadIdx.x) >> 5;
  int lane = threadIdx.x & 31;
  if (w >= n) return;
  float s = Hid[(size_t)w * 64 + lane] * W2[lane]
          + Hid[(size_t)w * 64 + 32 + lane] * W2[32 + lane];
  for (int off = 16; off > 0; off >>= 1) s += __shfl_down(s, off);
  if (lane == 0) g[w] = s + b2[0];
}

// segment max via signed-max / unsigned-min atomic punning (gmax init -inf)
__global__ void k_seg_max(const float* __restrict__ g, const int* __restrict__ batch,
                          float* __restrict__ gmax, int n) {
  int i = blockIdx.x * blockDim.x + threadIdx.x;
  if (i >= n) return;
  float v = g[i]; int b = batch[i];
  if (v >= 0.0f) atomicMax((int*)&gmax[b], __float_as_int(v));
  else           atomicMin((unsigned int*)&gmax[b], __float_as_uint(v));
}

__global__ void k_seg_exp(const float* __restrict__ g, const int* __restrict__ batch,
                          const float* __restrict__ gmax, float* __restrict__ e,
                          float* __restrict__ denom, int n) {
  int i = blockIdx.x * blockDim.x + threadIdx.x;
  if (i >= n) return;
  int b = batch[i];
  float v = expf(g[i] - gmax[b]);
  e[i] = v;
  atomicAdd(&denom[b], v);
}

// cat[b, off+0:64] += (e/denom) * x[n, 0:64]   (warp per node)
__global__ void k_att_out(const float* __restrict__ x, const int* __restrict__ batch,
                          const float* __restrict__ e, const float* __restrict__ denom,
                          float* __restrict__ cat, int n, int off) {
  int w = (blockIdx.x * blockDim.x + threadIdx.x) >> 5;
  int lane = threadIdx.x & 31;
  if (w >= n) return;
  int b = batch[w];
  float a = e[w] / denom[b];
  atomicAdd(&cat[(size_t)b * 320 + off + lane],      a * x[(size_t)w * 64 + lane]);
  atomicAdd(&cat[(size_t)b * 320 + off + 32 + lane], a * x[(size_t)w * 64 + 32 + lane]);
}

__global__ void k_count_deg(const int* __restrict__ dst, float* __restrict__ deg, int E_) {
  int i = blockIdx.x * blockDim.x + threadIdx.x;
  if (i < E_) atomicAdd(&deg[dst[i]], 1.0f);
}

__global__ void k_rsqrt(float* __restrict__ d, int n) {
  int i = blockIdx.x * blockDim.x + threadIdx.x;
  if (i < n) d[i] = rsqrtf(d[i]);
}

// Y[n,f] = bias[f] + dinv[n]^2 * Xw[n,f]   (self-loop term + bias)
__global__ void k_agg_init(const float* __restrict__ Xw, const float* __restrict__ dinv,
                           const float* __restrict__ bias, float* __restrict__ Y,
                           int n, int H_) {
  int i = blockIdx.x * blockDim.x + threadIdx.x;
  if (i >= n * H_) return;
  int node = i / H_, f = i - node * H_;
  float di = dinv[node];
  Y[i] = bias[f] + di * di * Xw[i];
}

// warp-per-edge scatter: Y[dst] += dinv[s]*dinv[d] * X[src]   (H floats)
__global__ void k_scatter_edges(const int* __restrict__ src, const int* __restrict__ dst,
                                const float* __restrict__ dinv, const float* __restrict__ X,
                                float* __restrict__ Y, int E_, int H_) {
  int w = (blockIdx.x * blockDim.x + threadIdx.x) >> 5;
  int lane = threadIdx.x & 31;
  int nw = (gridDim.x * blockDim.x) >> 5;
  for (int eix = w; eix < E_; eix += nw) {
    int s = src[eix], d = dst[eix];
    float wt = dinv[s] * dinv[d];
    for (int f = lane; f < H_; f += 32)
      atomicAdd(&Y[(size_t)d * H_ + f], wt * X[(size_t)s * H_ + f]);
  }
}

__global__ void k_mean_acc(const float* __restrict__ X, const int* __restrict__ batch,
                           float* __restrict__ cat, float* __restrict__ cnt,
                           int n, int off) {
  int w = (blockIdx.x * blockDim.x + threadIdx.x) >> 5;
  int lane = threadIdx.x & 31;
  if (w >= n) return;
  int b = batch[w];
  if (lane == 0) atomicAdd(&cnt[b], 1.0f);
  for (int f = lane; f < 96; f += 32)
    atomicAdd(&cat[(size_t)b * 320 + off + f], X[(size_t)w * 96 + f]);
}

__global__ void k_mean_div(float* __restrict__ cat, const float* __restrict__ cnt,
                           int nb, int off) {
  int i = blockIdx.x * blockDim.x + threadIdx.x;
  if (i >= nb * 96) return;
  int b = i / 96, f = i - b * 96;
  cat[(size_t)b * 320 + off + f] /= fmaxf(cnt[b], 1.0f);
}

// out[b] = dot(H1[b,0:96], W) + bias   (warp per row)
__global__ void k_final_dot(const float* __restrict__ H1, const float* __restrict__ W,
                            const float* __restrict__ b, float* __restrict__ out, int nb) {
  int w = (blockIdx.x * blockDim.x + threadIdx.x) >> 5;
  int lane = threadIdx.x & 31;
  if (w >= nb) return;
  float s = 0.0f;
  for (int f = lane; f < 96; f += 32) s += H1[(size_t)w * 96 + f] * W[f];
  for (int off = 16; off > 0; off >>= 1) s += __shfl_down(s, off);
  if (lane == 0) out[w] = s + b[0];
}

// ---------------------------------------------------------------------------

static inline void gemm_launch(const float* X, const float* W, const float* bias,
                               float* Y, int M, int Kin, int Kout,
                               int relu_in, int relu_out, int use_bias,
                               hipStream_t stream) {
  dim3 grid((M + 15) / 16);
  dim3 block(32 * (Kout / 16));
  size_t shmem = (size_t)16 * (Kin + 1) * sizeof(float);
  k_gemm_wmma<<<grid, block, shmem, stream>>>(X, W, bias, Y, M, Kin, Kout,
                                              relu_in, relu_out, use_bias);
}

extern "C" void kernel_launch(void* const* d_in, const int* in_sizes, int n_in,
                              void* d_out, int out_size, void* d_ws, size_t ws_size,
                              hipStream_t stream) {
  (void)n_in; (void)ws_size;
  const int N_ = in_sizes[0] / 64;   // nodes
  const int E_ = in_sizes[4] / 2;    // edges
  const int B_ = out_size;           // graphs

  // ---- workspace layout (floats) ----
  float* ws = (float*)d_ws;
  size_t o = 0;
  float* bufA  = ws + o; o += (size_t)N_ * 96;
  float* bufB  = ws + o; o += (size_t)N_ * 96;
  float* g     = ws + o; o += (size_t)N_;
  float* eacc  = ws + o; o += (size_t)N_;
  float* dinv  = ws + o; o += (size_t)N_;
  float* gmax  = ws + o; o += (size_t)B_;
  float* denom = ws + o; o += (size_t)B_;
  float* cnt   = ws + o; o += (size_t)B_;
  float* cat   = ws + o; o += (size_t)B_ * 320;
  float* h1    = ws + o; o += (size_t)B_ * 96;

  const float* gate_W1 = (const float*)d_in[8];
  const float* gate_b1 = (const float*)d_in[9];
  const float* gate_W2 = (const float*)d_in[10];
  const float* gate_b2 = (const float*)d_in[11];

  auto blocks = [](long long n, int t) { return dim3((unsigned)((n + t - 1) / t)); };

  // zero the concat buffer (att_out/mean_acc accumulate into it)
  k_fill<<<blocks((long long)B_ * 320, TPB), TPB, 0, stream>>>(cat, 0.0f, B_ * 320);

  for (int side = 0; side < 2; ++side) {
    const float* x     = (const float*)d_in[side];
    const int*   ei    = (const int*)d_in[4 + side];
    const int*   batch = (const int*)d_in[6 + side];
    const int*   src   = ei;
    const int*   dst   = ei + E_;
    const int    cb    = 12 + side * 6;  // conv weight base index
    const float* W0 = (const float*)d_in[cb + 0]; const float* b0 = (const float*)d_in[cb + 1];
    const float* W1 = (const float*)d_in[cb + 2]; const float* b1 = (const float*)d_in[cb + 3];
    const float* W2 = (const float*)d_in[cb + 4]; const float* b2 = (const float*)d_in[cb + 5];

    // ---- attention pool (gate MLP -> segment softmax -> weighted sum) ----
    gemm_launch(x, gate_W1, gate_b1, bufA, N_, 64, 64, 0, 1, 1, stream);       // relu(x@W1+b1)
    k_gate_g<<<blocks((long long)N_ * 32, TPB), TPB, 0, stream>>>(bufA, gate_W2, gate_b2, g, N_);
    k_fill<<<blocks(B_, TPB), TPB, 0, stream>>>(gmax, -__builtin_inff(), B_);
    k_fill<<<blocks(B_, TPB), TPB, 0, stream>>>(denom, 0.0f, B_);
    k_seg_max<<<blocks(N_, TPB), TPB, 0, stream>>>(g, batch, gmax, N_);
    k_seg_exp<<<blocks(N_, TPB), TPB, 0, stream>>>(g, batch, gmax, eacc, denom, N_);
    k_att_out<<<blocks((long long)N_ * 32, TPB), TPB, 0, stream>>>(
        x, batch, eacc, denom, cat, N_, 192 + side * 64);

    // ---- GCN normalization: dinv = rsqrt(deg + selfloop) ----
    k_fill<<<blocks(N_, TPB), TPB, 0, stream>>>(dinv, 1.0f, N_);               // self loop
    k_count_deg<<<blocks(E_, TPB), TPB, 0, stream>>>(dst, dinv, E_);
    k_rsqrt<<<blocks(N_, TPB), TPB, 0, stream>>>(dinv, N_);

    // ---- conv0: x[N,64] @ W0[64,96] -> aggregate ----
    gemm_launch(x, W0, nullptr, bufA, N_, 64, 96, 0, 0, 0, stream);
    k_agg_init<<<blocks((long long)N_ * 96, TPB), TPB, 0, stream>>>(bufA, dinv, b0, bufB, N_, 96);
    k_scatter_edges<<<4096, TPB, 0, stream>>>(src, dst, dinv, bufA, bufB, E_, 96);

    // ---- conv1: relu(prev) @ W1[96,96] -> aggregate ----
    gemm_launch(bufB, W1, nullptr, bufA, N_, 96, 96, 1, 0, 0, stream);
    k_agg_init<<<blocks((long long)N_ * 96, TPB), TPB, 0, stream>>>(bufA, dinv, b1, bufB, N_, 96);
    k_scatter_edges<<<4096, TPB, 0, stream>>>(src, dst, dinv, bufA, bufB, E_, 96);

    // ---- conv2: relu(prev) @ W2[96,96] -> aggregate (no trailing relu) ----
    gemm_launch(bufB, W2, nullptr, bufA, N_, 96, 96, 1, 0, 0, stream);
    k_agg_init<<<blocks((long long)N_ * 96, TPB), TPB, 0, stream>>>(bufA, dinv, b2, bufB, N_, 96);
    k_scatter_edges<<<4096, TPB, 0, stream>>>(src, dst, dinv, bufA, bufB, E_, 96);

    // ---- mean pool into cat[:, side*96 : side*96+96] ----
    k_fill<<<blocks(B_, TPB), TPB, 0, stream>>>(cnt, 0.0f, B_);
    k_mean_acc<<<blocks((long long)N_ * 32, TPB), TPB, 0, stream>>>(bufB, batch, cat, cnt, N_, side * 96);
    k_mean_div<<<blocks((long long)B_ * 96, TPB), TPB, 0, stream>>>(cat, cnt, B_, side * 96);
  }

  // ---- final MLP: relu(cat[B,320] @ lin_W0 + b0) @ lin_W1 + b1 ----
  gemm_launch(cat, (const float*)d_in[24], (const float*)d_in[25], h1,
              B_, 320, 96, 0, 1, 1, stream);
  k_final_dot<<<dim3((unsigned)(((long long)B_ * 32 + TPB - 1) / TPB)), TPB, 0, stream>>>(
      h1, (const float*)d_in[26], (const float*)d_in[27], (float*)d_out, B_);
}